// UHGTripletLoss_21328807592551
// MI455X (gfx1250) — compile-verified
//
#include <hip/hip_runtime.h>
#include <hip/hip_bf16.h>
#include <math.h>

typedef __attribute__((ext_vector_type(2))) float v2f;
typedef __attribute__((ext_vector_type(4))) float v4f;
typedef __attribute__((ext_vector_type(8))) float v8f;

#define N_ROWS      1048576
#define DDIM        128
#define TILE_M      16
#define LDS_STRIDE  132          // 128 + 4 floats pad: keeps 16B align, shifts banks by 4
#define WPB         2            // waves per block (64 threads) -> 50,688 B static LDS
#define BLOCKS      4096
#define NUM_TILES   (N_ROWS / TILE_M)   // 65536
#define TOTAL_WAVES (BLOCKS * WPB)      // 8192 persistent waves, 8 tiles each
#define MARGIN_F    1.0f
#define EPS_F       1e-6f

#define WMMA_F32(a, b, c) \
    __builtin_amdgcn_wmma_f32_16x16x4_f32(false, (a), false, (b), (short)0, (c), false, false)

__device__ __forceinline__ float sel8(v8f v, int i) {
    float r = v[0];
    r = (i == 1) ? v[1] : r;
    r = (i == 2) ? v[2] : r;
    r = (i == 3) ? v[3] : r;
    r = (i == 4) ? v[4] : r;
    r = (i == 5) ? v[5] : r;
    r = (i == 6) ? v[6] : r;
    r = (i == 7) ? v[7] : r;
    return r;
}

__device__ __forceinline__ float uhg_dist(float xy, float xx, float yy) {
    float ratio = (xy * xy) / fmaxf(fabsf(xx * yy), EPS_F);
    float s = sqrtf(fmaxf(ratio, 1.0f + EPS_F));
    return acoshf(s);
}

__global__ __launch_bounds__(WPB * 32)
void uhg_triplet_main(const float* __restrict__ A,
                      const float* __restrict__ P,
                      const float* __restrict__ Ng,
                      float* __restrict__ out_pos,
                      float* __restrict__ out_neg,
                      float* __restrict__ partials) {
    __shared__ float lds[WPB][3][TILE_M * LDS_STRIDE];
    __shared__ float s_wave[WPB];

    const int tid  = threadIdx.x;
    const int w    = tid >> 5;
    const int lane = tid & 31;
    const int lo16 = lane & 15;
    const int hi   = lane >> 4;            // 0: K+{0,1}, 1: K+{2,3}

    float* la = lds[w][0];
    float* lp = lds[w][1];
    float* ln = lds[w][2];

    float lossPartial = 0.0f;

    const int gw = blockIdx.x * WPB + w;
    for (int tile = gw; tile < NUM_TILES; tile += TOTAL_WAVES) {
        const int baseRow = tile * TILE_M;

        // ---- stage 16x128 of a/p/n to LDS: fully coalesced 128-bit loads/stores ----
        for (int i = 0; i < TILE_M; ++i) {
            const size_t goff = (size_t)(baseRow + i) * DDIM + (size_t)(lane * 4);
            v4f va = *(const v4f*)(A  + goff);
            v4f vp = *(const v4f*)(P  + goff);
            v4f vn = *(const v4f*)(Ng + goff);
            const int loff = i * LDS_STRIDE + lane * 4;
            *(v4f*)(la + loff) = va;
            *(v4f*)(lp + loff) = vp;
            *(v4f*)(ln + loff) = vn;
        }
        // LDS region is wave-private: no barrier needed, only ds-counter waits.

        v8f c_ap = {}; v8f c_aa = {}; v8f c_pp = {}; v8f c_an = {}; v8f c_nn = {};

        // Per-lane operand offset: lane l supplies X[M = l&15][K = k + 2*(l>>4) + {0,1}],
        // which is simultaneously the A-operand (16x4) and the B-operand (4x16 = X^T) layout.
        const int opOff = lo16 * LDS_STRIDE + 2 * hi;

        for (int k = 0; k < DDIM; k += 4) {
            v2f oa = *(const v2f*)(la + opOff + k);
            v2f op = *(const v2f*)(lp + opOff + k);
            v2f on = *(const v2f*)(ln + opOff + k);

            // Minkowski form: negate column 127 on the B side only
            // (col 127 = component 1 of lanes>=16 at k == 124). Branch-free.
            const float sgn = (hi && (k == DDIM - 4)) ? -1.0f : 1.0f;
            v2f ba = oa; ba[1] *= sgn;
            v2f bp = op; bp[1] *= sgn;
            v2f bn = on; bn[1] *= sgn;

            c_ap = WMMA_F32(oa, bp, c_ap);   // diag -> <a,p>
            c_aa = WMMA_F32(oa, ba, c_aa);   // diag -> <a,a>
            c_pp = WMMA_F32(op, bp, c_pp);   // diag -> <p,p>
            c_an = WMMA_F32(oa, bn, c_an);   // diag -> <a,n>
            c_nn = WMMA_F32(on, bn, c_nn);   // diag -> <n,n>
        }

        // Diagonal of 16x16 f32 C: row i<8 lives in (lane i, vgpr i);
        // row i>=8 lives in (lane i+16, vgpr i-8). Only lanes 0-7 and 24-31 hold diag.
        const bool active    = (lane < 8) || (lane >= 24);
        const int  vi        = (lane < 8) ? lane : (lane - 24);
        const int  rowInTile = (lane < 8) ? lane : (lane - 16);
        if (active) {
            float ap = sel8(c_ap, vi);
            float aa = sel8(c_aa, vi);
            float pp = sel8(c_pp, vi);
            float an = sel8(c_an, vi);
            float nn = sel8(c_nn, vi);
            float dp = uhg_dist(ap, aa, pp);
            float dn = uhg_dist(an, aa, nn);
            const int row = baseRow + rowInTile;
            out_pos[row] = dp;
            out_neg[row] = dn;
            lossPartial += fmaxf(dp - dn + MARGIN_F, 0.0f);
        }
    }

    // deterministic wave tree-reduce, then per-block partial
    for (int off = 16; off > 0; off >>= 1)
        lossPartial += __shfl_down(lossPartial, off, 32);
    if (lane == 0) s_wave[w] = lossPartial;
    __syncthreads();
    if (tid == 0) {
        float s = 0.0f;
        for (int i = 0; i < WPB; ++i) s += s_wave[i];
        partials[blockIdx.x] = s;
    }
}

__global__ __launch_bounds__(256)
void uhg_triplet_reduce(const float* __restrict__ partials,
                        float* __restrict__ out_loss,
                        int nblocks, float invN) {
    __shared__ float s[256];
    float acc = 0.0f;
    for (int i = threadIdx.x; i < nblocks; i += 256) acc += partials[i];
    s[threadIdx.x] = acc;
    __syncthreads();
    for (int off = 128; off > 0; off >>= 1) {
        if (threadIdx.x < off) s[threadIdx.x] += s[threadIdx.x + off];
        __syncthreads();
    }
    if (threadIdx.x == 0) out_loss[0] = s[0] * invN;
}

extern "C" void kernel_launch(void* const* d_in, const int* in_sizes, int n_in,
                              void* d_out, int out_size, void* d_ws, size_t ws_size,
                              hipStream_t stream) {
    (void)in_sizes; (void)n_in; (void)out_size; (void)ws_size;
    const float* A  = (const float*)d_in[0];
    const float* P  = (const float*)d_in[1];
    const float* Ng = (const float*)d_in[2];

    float* out      = (float*)d_out;
    float* out_loss = out;                 // [1]
    float* out_pos  = out + 1;             // [N]
    float* out_neg  = out + 1 + N_ROWS;    // [N]
    float* partials = (float*)d_ws;        // BLOCKS floats of scratch

    uhg_triplet_main<<<BLOCKS, WPB * 32, 0, stream>>>(A, P, Ng, out_pos, out_neg, partials);
    uhg_triplet_reduce<<<1, 256, 0, stream>>>(partials, out_loss, BLOCKS,
                                              1.0f / (float)N_ROWS);
}